// SoftCluster_75462575391235
// MI455X (gfx1250) — compile-verified
//
#include <hip/hip_runtime.h>
#include <math.h>

// ---- problem constants ----
#define N_ROWS 20000
#define DIMS   1024
#define K_CL   5
#define K_PAD  16
#define ITERS  400
#define NSLAB  20
#define ROWS_PER_SLAB (N_ROWS / NSLAB)   // 1000 (multiple of 4)
#define DTILES (DIMS / 16)               // 64
#define DGROUPS (DTILES / 8)             // 8 (8 waves/block, 1 tile each)
#define BETA_C 1.0f
#define TOL_C  1e-5f
#define EPS_C  1e-12f

typedef __attribute__((ext_vector_type(2))) float v2f;
typedef __attribute__((ext_vector_type(8))) float v8f;
typedef __attribute__((ext_vector_type(4))) int   v4i;

// pointer-to-addrspace types for the async-to-LDS builtin:
// param0 = int4 addrspace(1)*, param1 = int4 addrspace(3)*
typedef __attribute__((address_space(1))) v4i* g_v4i_p;
typedef __attribute__((address_space(3))) v4i* s_v4i_p;

// ---- workspace layout (float offsets) ----
#define CENTERS_OFF 0                         // [16][1024]
#define PREV_OFF    16384                     // [16][1024]
#define CSQ_OFF     32768                     // [16]
#define XSQ_OFF     32784                     // [20000]
#define RPAD_OFF    52784                     // [20000][16]
#define CPART_OFF   372784                    // [NSLAB][5][1024]
#define DONE_OFF    475184                    // int flag

// Async-to-LDS staging (CDNA5): guarded so compile never breaks if the
// builtin names differ on this toolchain.
#if defined(__has_builtin)
#if __has_builtin(__builtin_amdgcn_global_load_async_to_lds_b128) && \
    __has_builtin(__builtin_amdgcn_s_wait_asynccnt)
#define USE_ASYNC_LDS 1
#endif
#endif

__device__ __forceinline__ int is_done(const float* ws) {
    return *(const volatile int*)(ws + DONE_OFF);
}

// Deterministic init: centers[k] = x[997 + 4001*k], pad rows 5..15 with 0.
__global__ void init_centers_kernel(const float* __restrict__ x, float* ws) {
    int idx = blockIdx.x * blockDim.x + threadIdx.x;
    if (idx == 0) *(int*)(ws + DONE_OFF) = 0;
    if (idx >= K_PAD * DIMS) return;
    int k = idx >> 10, d = idx & (DIMS - 1);
    float v = 0.0f;
    if (k < K_CL) {
        int row = 997 + 4001 * k;             // deterministic spread, < N_ROWS
        v = x[(size_t)row * DIMS + d];
    }
    ws[CENTERS_OFF + idx] = v;
    ws[PREV_OFF + idx] = 0.0f;
}

// ||c_k||^2 for initial centers (one block, 16 waves, wave per padded row)
__global__ void csq_kernel(float* ws) {
    int w = threadIdx.x >> 5, lane = threadIdx.x & 31;
    const float* c = ws + CENTERS_OFF + (size_t)w * DIMS;
    float s = 0.0f;
    for (int j = lane; j < DIMS; j += 32) { float v = c[j]; s += v * v; }
    for (int m = 16; m; m >>= 1) s += __shfl_xor(s, m, 32);
    if (lane == 0) ws[CSQ_OFF + w] = s;
}

// ||x_i||^2, wave per row (coalesced 128B/step)
__global__ void xsq_kernel(const float* __restrict__ x, float* ws) {
    int w = threadIdx.x >> 5, lane = threadIdx.x & 31;
    int row = blockIdx.x * 8 + w;
    const float* xr = x + (size_t)row * DIMS;
    float s = 0.0f;
    for (int j = lane; j < DIMS; j += 32) { float v = xr[j]; s += v * v; }
    for (int m = 16; m; m >>= 1) s += __shfl_xor(s, m, 32);
    if (lane == 0) ws[XSQ_OFF + row] = s;
}

// responsibilities: wave per row; centers staged in LDS (20KB); stable softmax
__global__ void resp_kernel(const float* __restrict__ x, float* ws) {
    if (is_done(ws)) return;
    __shared__ float cent[K_CL * DIMS];
    __shared__ float csq_s[K_CL];
    for (int i = threadIdx.x; i < K_CL * DIMS; i += 256)
        cent[i] = ws[CENTERS_OFF + i];
    if (threadIdx.x < K_CL) csq_s[threadIdx.x] = ws[CSQ_OFF + threadIdx.x];
    __syncthreads();

    int w = threadIdx.x >> 5, lane = threadIdx.x & 31;
    int row = blockIdx.x * 8 + w;
    const float* xr = x + (size_t)row * DIMS;
    float a0 = 0, a1 = 0, a2 = 0, a3 = 0, a4 = 0;
    for (int j = lane; j < DIMS; j += 32) {
        float v = xr[j];
        a0 += v * cent[j];
        a1 += v * cent[DIMS + j];
        a2 += v * cent[2 * DIMS + j];
        a3 += v * cent[3 * DIMS + j];
        a4 += v * cent[4 * DIMS + j];
    }
    for (int m = 16; m; m >>= 1) {
        a0 += __shfl_xor(a0, m, 32);
        a1 += __shfl_xor(a1, m, 32);
        a2 += __shfl_xor(a2, m, 32);
        a3 += __shfl_xor(a3, m, 32);
        a4 += __shfl_xor(a4, m, 32);
    }
    float dot[K_CL] = {a0, a1, a2, a3, a4};
    float xs = ws[XSQ_OFF + row];
    float l[K_CL], mx = -1e30f;
    #pragma unroll
    for (int k = 0; k < K_CL; k++) {
        float sq = xs + csq_s[k] - 2.0f * dot[k];
        l[k] = -BETA_C * sqrtf(fmaxf(sq, EPS_C));
        mx = fmaxf(mx, l[k]);
    }
    float e[K_CL], se = 0.0f;
    #pragma unroll
    for (int k = 0; k < K_CL; k++) { e[k] = expf(l[k] - mx); se += e[k]; }
    float inv = 1.0f / se;
    if (lane < K_PAD) {
        float r = (lane < K_CL) ? e[lane] * inv : 0.0f;  // all lanes have e[]
        ws[RPAD_OFF + (size_t)row * K_PAD + lane] = r;
    }
}

// c_partial[slab] += r^T @ x via V_WMMA_F32_16X16X4_F32.
// Block = 8 waves; r-slab (1000x16, 64KB) staged once into LDS (async if
// available), each wave owns one 16-column D-tile; K dimension = rows.
__global__ void accum_kernel(const float* __restrict__ x, float* ws) {
    if (is_done(ws)) return;
    __shared__ float rslab[ROWS_PER_SLAB * K_PAD];   // 64000 B

    int rowBase = blockIdx.y * ROWS_PER_SLAB;
    float* rp_slab = ws + RPAD_OFF + (size_t)rowBase * K_PAD;

#ifdef USE_ASYNC_LDS
    for (int i = threadIdx.x; i < (ROWS_PER_SLAB * K_PAD) / 4; i += 256) {
        __builtin_amdgcn_global_load_async_to_lds_b128(
            (g_v4i_p)(rp_slab + 4 * i), (s_v4i_p)(rslab + 4 * i), 0, 0);
    }
    __builtin_amdgcn_s_wait_asynccnt(0);
#else
    for (int i = threadIdx.x; i < (ROWS_PER_SLAB * K_PAD) / 4; i += 256) {
        ((float4*)rslab)[i] = ((const float4*)rp_slab)[i];
    }
#endif
    __syncthreads();

    int w = threadIdx.x >> 5;
    int lane = threadIdx.x & 31;
    int n  = lane & 15;      // N index (D-col for B/D, cluster for A)
    int hi = lane >> 4;      // half-wave selector -> K pair {0,1} vs {2,3}
    int dt = blockIdx.x * 8 + w;          // this wave's D-tile
    int dBase = dt * 16;

    v8f acc = {};
    for (int c = 0; c < ROWS_PER_SLAB; c += 4) {
        int l0 = c + 2 * hi;              // local row in slab
        int r0 = rowBase + l0;            // global row
        v2f a, b;
        // A (16x4, A[m][k] = r[row_k][cluster m]) from LDS:
        a.x = rslab[l0 * K_PAD + n];
        a.y = rslab[(l0 + 1) * K_PAD + n];
        // B (4x16, B[k][n] = x[row_k][dBase+n]) from global:
        b.x = x[(size_t)r0 * DIMS + dBase + n];
        b.y = x[(size_t)(r0 + 1) * DIMS + dBase + n];
        acc = __builtin_amdgcn_wmma_f32_16x16x4_f32(
            /*neg_a=*/false, a, /*neg_b=*/false, b,
            /*c_mod=*/(short)0, acc, /*reuse_a=*/false, /*reuse_b=*/false);
    }
    // D layout: element (m = v + 8*hi, n); only clusters m<5 are real.
    float* cp = ws + CPART_OFF + (size_t)blockIdx.y * K_CL * DIMS + dBase;
    #pragma unroll
    for (int v = 0; v < 8; v++) {
        int m = v + 8 * hi;
        if (m < K_CL) cp[(size_t)m * DIMS + n] = acc[v];
    }
}

// single block: reduce rsum + slab partials, divide, shift, update state
__global__ void finalize_kernel(float* ws) {
    if (is_done(ws)) return;
    __shared__ float red[512];
    __shared__ float rsum_s[K_PAD];
    int tid = threadIdx.x;

    // 1) rsum (column sums of r_pad), deterministic tree reduce
    int col = tid & 15, seg = tid >> 4;   // 32 segments x 16 cols
    float s = 0.0f;
    for (int row = seg; row < N_ROWS; row += 32)
        s += ws[RPAD_OFF + (size_t)row * K_PAD + col];
    red[tid] = s; __syncthreads();
    for (int h = 16; h >= 1; h >>= 1) {
        if (seg < h) red[tid] += red[(seg + h) * 16 + col];
        __syncthreads();
    }
    if (tid < K_PAD) rsum_s[tid] = red[tid];
    __syncthreads();

    // 2) c_new = sum_slabs(partial) / rsum ; accumulate shift vs prev
    float sh = 0.0f;
    for (int idx = tid; idx < K_CL * DIMS; idx += 512) {
        int k = idx >> 10;
        float sum = 0.0f;
        for (int sl = 0; sl < NSLAB; sl++)
            sum += ws[CPART_OFF + (size_t)sl * K_CL * DIMS + idx];
        float v = sum / fmaxf(rsum_s[k], 1e-30f);
        float d = v - ws[PREV_OFF + idx];
        sh += d * d;
        ws[CENTERS_OFF + idx] = v;   // rows 5..15 stay 0 from init
        ws[PREV_OFF + idx] = v;
    }
    __syncthreads();
    red[tid] = sh; __syncthreads();
    for (int h = 256; h >= 1; h >>= 1) {
        if (tid < h) red[tid] += red[tid + h];
        __syncthreads();
    }
    if (tid == 0) {
        float shift = sqrtf(red[0]);
        if (shift < TOL_C) *(int*)(ws + DONE_OFF) = 1;
    }
    __syncthreads();

    // 3) refresh ||c_k||^2 for next iteration (wave per padded row)
    int w = tid >> 5, lane = tid & 31;
    const float* c = ws + CENTERS_OFF + (size_t)w * DIMS;
    float cs = 0.0f;
    for (int j = lane; j < DIMS; j += 32) { float vv = c[j]; cs += vv * vv; }
    for (int m = 16; m; m >>= 1) cs += __shfl_xor(cs, m, 32);
    if (lane == 0) ws[CSQ_OFF + w] = cs;
}

// pack outputs: centers [5,1024] then r [20000,5]
__global__ void output_kernel(const float* __restrict__ ws, float* __restrict__ out) {
    int idx = blockIdx.x * 256 + threadIdx.x;
    if (idx < K_CL * DIMS) {
        out[idx] = ws[CENTERS_OFF + idx];
    } else {
        int j = idx - K_CL * DIMS;
        if (j < N_ROWS * K_CL) {
            int row = j / K_CL, k = j - row * K_CL;
            out[idx] = ws[RPAD_OFF + (size_t)row * K_PAD + k];
        }
    }
}

extern "C" void kernel_launch(void* const* d_in, const int* in_sizes, int n_in,
                              void* d_out, int out_size, void* d_ws, size_t ws_size,
                              hipStream_t stream) {
    (void)in_sizes; (void)n_in; (void)out_size; (void)ws_size;
    const float* x = (const float*)d_in[0];
    float* ws = (float*)d_ws;
    float* out = (float*)d_out;

    init_centers_kernel<<<(K_PAD * DIMS + 255) / 256, 256, 0, stream>>>(x, ws);
    csq_kernel<<<1, 512, 0, stream>>>(ws);
    xsq_kernel<<<N_ROWS / 8, 256, 0, stream>>>(x, ws);

    for (int t = 0; t < ITERS; t++) {
        resp_kernel<<<N_ROWS / 8, 256, 0, stream>>>(x, ws);
        accum_kernel<<<dim3(DGROUPS, NSLAB), 256, 0, stream>>>(x, ws);
        finalize_kernel<<<1, 512, 0, stream>>>(ws);
    }

    int total = K_CL * DIMS + N_ROWS * K_CL;          // 105120
    output_kernel<<<(total + 255) / 256, 256, 0, stream>>>(ws, out);
}